// IntGELU_3393024163895
// MI455X (gfx1250) — compile-verified
//
#include <hip/hip_runtime.h>
#include <stdint.h>

// IntGELU (I-BERT style integer GELU) for MI455X / gfx1250.
// Memory-bound floor: ~310 MB traffic -> ~13.4 us at 23.3 TB/s.
// VALU budget trimmed to match: hoisted reciprocal for the uniform divisor,
// rcp+Newton for the per-element divide (full IEEE div was ~10 VALU/elem x2).
// CDNA5 paths: global_load_async_to_lds_b128 + s_wait_asynccnt staging,
// wave32 shuffle reduction, 128-bit non-temporal stores.

typedef float f4 __attribute__((ext_vector_type(4)));

#define ROW_LEN 3072
#define VECS_PER_ROW (ROW_LEN / 4)   // 768 float4 per row
#define THREADS 256                  // 8 wave32
#define VPT (VECS_PER_ROW / THREADS) // 3 float4 per thread
#define NWAVES (THREADS / 32)

// ~0.5 ulp reciprocal: v_rcp_f32 + one Newton-Raphson step (2 fma + 1 trans)
__device__ __forceinline__ float fast_rcp(float x) {
#if defined(__gfx1250__) && __has_builtin(__builtin_amdgcn_rcpf)
    float r = __builtin_amdgcn_rcpf(x);
    return fmaf(fmaf(-x, r, 1.0f), r, r);
#else
    return 1.0f / x;
#endif
}

// int_exp_shift with uniform scalars precomputed:
//   x0 = floor(-1/sf_sig), rq = 1/x0, clampv = 23*x0
__device__ __forceinline__ float int_exp_shift(float v, float x0, float rq,
                                               float clampv) {
    float t = v + floorf(v * 0.5f) - floorf(v * 0.0625f);
    t = fmaxf(t, clampv);
    float q = floorf(t * rq);        // t/x0 via hoisted reciprocal
    float r = fmaf(-x0, q, t);       // remainder
    float e = fmaf(r, 0.5f, -x0);
    return fmaxf(floorf(ldexpf(e, 23 - (int)q)), 0.0f);  // v_ldexp_f32
}

__global__ __launch_bounds__(THREADS) void intgelu_kernel(
    const float* __restrict__ x, const float* __restrict__ sfp,
    float* __restrict__ out, int rows)
{
    __shared__ f4 tile4[VECS_PER_ROW];       // 12 KB row tile
    __shared__ float partials[NWAVES];

    const int row = blockIdx.x;
    const int tid = threadIdx.x;

    const float s      = sfp[0];
    const float inv_s  = 1.0f / s;
    const float sf_sig = s * 1.702f;
    const float x0     = floorf(-1.0f / sf_sig);   // integer -ln2 analogue
    const float rq     = 1.0f / x0;                // hoisted: divide once
    const float clampv = 23.0f * x0;

    const f4* rowp = (const f4*)(x + (size_t)row * ROW_LEN);

#if defined(__gfx1250__)
    // CDNA5 async copy engine: global -> LDS, 128b per lane, ASYNCcnt-tracked.
    // Each lane loads exactly the LDS words it reads back, so a per-wave
    // s_wait_asynccnt 0 is sufficient for the tile (no cross-wave barrier).
    #pragma unroll
    for (int k = 0; k < VPT; ++k) {
        const int idx = tid + THREADS * k;
        const unsigned goff  = (unsigned)idx * 16u;                  // byte offset
        const unsigned laddr = (unsigned)(uintptr_t)&tile4[idx];     // LDS address
        asm volatile("global_load_async_to_lds_b128 %0, %1, %2"
                     :: "v"(laddr), "v"(goff), "s"(rowp) : "memory");
    }
    asm volatile("s_wait_asynccnt 0x0" ::: "memory");
#else
    #pragma unroll
    for (int k = 0; k < VPT; ++k) {
        const int idx = tid + THREADS * k;
        tile4[idx] = rowp[idx];
    }
#endif

    // Pass 1: per-thread max over pre_x_int = x/s (kept in regs for pass 2).
    f4 pre[VPT];
    float m = -3.402823466e38f;
    #pragma unroll
    for (int k = 0; k < VPT; ++k) {
        const int idx = tid + THREADS * k;
        f4 v = tile4[idx];                 // ds_load_b128
        v *= inv_s;
        pre[k] = v;
        m = fmaxf(m, fmaxf(fmaxf(v.x, v.y), fmaxf(v.z, v.w)));  // v_max3
    }

    // wave32 xor-shuffle max tree, then cross-wave combine via LDS
    #pragma unroll
    for (int off = 16; off >= 1; off >>= 1)
        m = fmaxf(m, __shfl_xor(m, off, 32));
    if ((tid & 31) == 0) partials[tid >> 5] = m;
    __syncthreads();
    #pragma unroll
    for (int w = 0; w < NWAVES; ++w) m = fmaxf(m, partials[w]);

    const float e_m       = int_exp_shift(-m, x0, rq, clampv);  // exp(-max)
    const float out_scale = s * 0.0078125f;                     // s / 2^7

    f4* orow = (f4*)(out + (size_t)row * ROW_LEN);
    #pragma unroll
    for (int k = 0; k < VPT; ++k) {
        const int idx = tid + THREADS * k;
        const f4 p = pre[k];
        f4 o;
        #pragma unroll
        for (int j = 0; j < 4; ++j) {
            const float pj     = p[j];
            const float e      = int_exp_shift(pj - m, x0, rq, clampv);
            const float sum    = fminf(e + e_m, 2147483647.0f);
            const float factor = floorf(2147483647.0f * fast_rcp(sum));
            const float sig    = floorf(e * factor * 0x1p-24f);  // / 2^24
            o[j] = pj * sig * out_scale;
        }
        // output never re-read: 128-bit non-temporal store, keep L2 clean
        __builtin_nontemporal_store(o, &orow[idx]);
    }

    // Tuple output: second return value (out_sf scalar) appended at the end.
    if (row == 0 && tid == 0) out[(size_t)rows * ROW_LEN] = out_scale;
}

extern "C" void kernel_launch(void* const* d_in, const int* in_sizes, int n_in,
                              void* d_out, int out_size, void* d_ws, size_t ws_size,
                              hipStream_t stream) {
    (void)n_in; (void)out_size; (void)d_ws; (void)ws_size;
    const float* x  = (const float*)d_in[0];
    const float* sf = (const float*)d_in[1];
    float* out      = (float*)d_out;
    const int rows  = in_sizes[0] / ROW_LEN;   // 64*197 = 12608
    intgelu_kernel<<<dim3(rows), dim3(THREADS), 0, stream>>>(x, sf, out, rows);
}